// Model_6605659701438
// MI455X (gfx1250) — compile-verified
//
#include <hip/hip_runtime.h>
#include <hip/hip_bf16.h>
#include <stdint.h>

// Problem constants (from reference)
#define N_CLIPS   1024
#define R_TUBES   150
#define D4        64          // floats per tube row (16 boxes * 4)
#define KTOP      500
#define NEGV      (-10000.0f)
#define THRESH    0.5f

// Workspace layout (bytes)
#define BP_STRIDE 152                         // u8 backptrs per step, padded
#define BP_OFF    0                           // u8 bp[1023][152]      = 155,496 B
#define SC_OFF    155648                      // float scores[150]     =     600 B
#define EI_OFF    156672                      // int   end_idx[500]    =   2,000 B
#define POS_OFF   158720                      // u8    pos[500*1024]   = 512,000 B

// ---------------------------------------------------------------------------
// CDNA5 Tensor Data Mover path (guarded; fallback = plain vector loads)
// ---------------------------------------------------------------------------
#if defined(__gfx1250__) && __has_builtin(__builtin_amdgcn_tensor_load_to_lds) && \
    __has_builtin(__builtin_amdgcn_s_wait_tensorcnt)
#define USE_TDM 1
#else
#define USE_TDM 0
#endif

typedef unsigned int u32x4 __attribute__((ext_vector_type(4)));
typedef int          i32x4 __attribute__((ext_vector_type(4)));
typedef int          i32x8 __attribute__((ext_vector_type(8)));

#if USE_TDM
// DMA a [rows x tileW] tile of f32 (row stride strideElems, in elements) from
// global memory into contiguous LDS at ldst.  2D D# per CDNA5 ISA: group0
// carries count/lds_addr/global_addr/type=2, group1 carries dims/strides.
// This toolchain exposes the 6-arg builtin: (g0, g1, g2, g3, g_extra, cpol);
// groups 2/3/extra are zero for a 2D tile.
__device__ __forceinline__ void tdm_load_tile_f32(const void* gsrc, void* ldst,
                                                  unsigned tileW, unsigned rows,
                                                  unsigned strideElems) {
  unsigned long long ga = (unsigned long long)gsrc;
  unsigned lds = (unsigned)(unsigned long long)ldst;  // low 32b of flat shared ptr == LDS byte offset
  u32x4 g0 = { 1u,                                    // count=1, user descriptor
               lds,                                   // [63:32]  lds_addr
               (unsigned)ga,                          // [95:64]  global_addr lo
               ((unsigned)(ga >> 32) & 0x01FFFFFFu)   // [120:96] global_addr hi
                 | 0x80000000u };                     // [127:126] type=2
  unsigned w0 = 2u << 16;                             // data_size=4B; mask/flags=0
  unsigned w1 = (tileW & 0xFFFFu) << 16;              // tensor_dim0 lo16 @ [63:48]
  unsigned w2 = ((tileW >> 16) & 0xFFFFu) | ((rows & 0xFFFFu) << 16);   // td0 hi | td1 lo
  unsigned w3 = ((rows >> 16) & 0xFFFFu) | ((tileW & 0xFFFFu) << 16);   // td1 hi | tile_dim0
  unsigned w4 = rows & 0xFFFFu;                       // tile_dim1 | tile_dim2=0
  unsigned w5 = strideElems;                          // tensor_dim0_stride lo32
  i32x8 g1 = { (int)w0, (int)w1, (int)w2, (int)w3, (int)w4, (int)w5, 0, 0 };
  i32x4 gz4 = { 0, 0, 0, 0 };
  i32x8 gz8 = { 0, 0, 0, 0, 0, 0, 0, 0 };
  __builtin_amdgcn_tensor_load_to_lds(g0, g1, gz4, gz4, gz8, 0);
}
#endif

// ---------------------------------------------------------------------------
// Phase 1: sequential max-plus Viterbi forward pass. One workgroup, 640
// threads: 4 lanes per destination column j, ping-pong score vectors in LDS.
// TDM double-buffers next step's two 150x4 f32 tiles (stride-64 gathers).
// ---------------------------------------------------------------------------
__global__ void __launch_bounds__(640)
viterbi_fwd(const float* __restrict__ p_tubes, const float* __restrict__ act_g,
            unsigned char* __restrict__ bp, float* __restrict__ scores_out) {
  __shared__ float boxPrev[2][R_TUBES * 4];
  __shared__ float boxCur [2][R_TUBES * 4];
  __shared__ float actL[R_TUBES];
  __shared__ float scoreL[2][R_TUBES];
  __shared__ float pmax[R_TUBES][4];
  __shared__ int   pidx[R_TUBES][4];

  const int tid = threadIdx.x;

  if (tid < R_TUBES) scoreL[0][tid] = act_g[tid];   // carry init = actioness[0]

#if USE_TDM
  if (tid < 32) {  // wave 0 issues DMA; TDM ignores EXEC, one issue per wave
    tdm_load_tile_f32(p_tubes + 60,                    boxPrev[1], 4, R_TUBES, D4);
    tdm_load_tile_f32(p_tubes + (size_t)R_TUBES * D4,  boxCur [1], 4, R_TUBES, D4);
  }
#endif

  for (int t = 1; t < N_CLIPS; ++t) {
    const int b = t & 1;
    const int sp = (t - 1) & 1;

#if USE_TDM
    if (tid < 32) {
      if (t + 1 < N_CLIPS) {
        // prefetch step t+1 into the other buffer, retire step t's pair
        tdm_load_tile_f32(p_tubes + ((size_t)t       * R_TUBES) * D4 + 60, boxPrev[b ^ 1], 4, R_TUBES, D4);
        tdm_load_tile_f32(p_tubes + ((size_t)(t + 1) * R_TUBES) * D4,      boxCur [b ^ 1], 4, R_TUBES, D4);
        __builtin_amdgcn_s_wait_tensorcnt(2);
      } else {
        __builtin_amdgcn_s_wait_tensorcnt(0);
      }
    }
#else
    if (tid < 600) {
      int j = tid >> 2, c = tid & 3;
      boxPrev[b][j * 4 + c] = p_tubes[((size_t)(t - 1) * R_TUBES + j) * D4 + 60 + c];
      boxCur [b][j * 4 + c] = p_tubes[((size_t)t       * R_TUBES + j) * D4 + c];
    }
#endif
    if (tid < R_TUBES) actL[tid] = act_g[(size_t)t * R_TUBES + tid];
    __syncthreads();

    if (tid < 600) {
      const int j = tid >> 2, sub = tid & 3;
      const float bx1 = boxCur[b][j * 4 + 0], by1 = boxCur[b][j * 4 + 1];
      const float bx2 = boxCur[b][j * 4 + 2], by2 = boxCur[b][j * 4 + 3];
      const float areaB = (bx2 - bx1) * (by2 - by1);
      int i0 = sub * 38;
      int i1 = i0 + 38; if (i1 > R_TUBES) i1 = R_TUBES;
      float best = -3.0e38f; int bi = i0;
      for (int i = i0; i < i1; ++i) {
        const float ax1 = boxPrev[b][i * 4 + 0], ay1 = boxPrev[b][i * 4 + 1];
        const float ax2 = boxPrev[b][i * 4 + 2], ay2 = boxPrev[b][i * 4 + 3];
        const float w = fmaxf(fminf(ax2, bx2) - fmaxf(ax1, bx1), 0.0f);
        const float h = fmaxf(fminf(ay2, by2) - fmaxf(ay1, by1), 0.0f);
        const float inter = w * h;
        const float areaA = (ax2 - ax1) * (ay2 - ay1);
        const float iou = inter / (areaA + areaB - inter + 1e-8f);
        const float ov = (iou >= THRESH) ? iou : NEGV;
        const float cand = scoreL[sp][i] + ov;
        if (cand > best) { best = cand; bi = i; }  // strict > keeps first max
      }
      pmax[j][sub] = best;
      pidx[j][sub] = bi;
    }
    __syncthreads();

    if (tid < R_TUBES) {
      float best = pmax[tid][0]; int bi = pidx[tid][0];
#pragma unroll
      for (int s = 1; s < 4; ++s) {
        const float v = pmax[tid][s];
        if (v > best) { best = v; bi = pidx[tid][s]; }   // ascending subs => first-index ties
      }
      bp[(size_t)(t - 1) * BP_STRIDE + tid] = (unsigned char)bi;
      scoreL[b][tid] = best + actL[tid];
    }
    __syncthreads();   // protects score ping-pong + LDS tile reuse
  }

  if (tid < R_TUBES) scores_out[tid] = scoreL[(N_CLIPS - 1) & 1][tid];
}

// ---------------------------------------------------------------------------
// Phase 2: descending rank-sort of 150 scores (stable: ties -> lower index),
// emit top-K scores directly into the output tail + end indices.  K > n, so
// ranks >= 150 are deterministic padding.
// ---------------------------------------------------------------------------
__global__ void __launch_bounds__(256)
topk_sort(const float* __restrict__ scores, float* __restrict__ topk_out,
          int* __restrict__ end_idx) {
  __shared__ float s[R_TUBES];
  const int tid = threadIdx.x;
  if (tid < R_TUBES) s[tid] = scores[tid];
  __syncthreads();
  if (tid < R_TUBES) {
    const float v = s[tid];
    int rank = 0;
    for (int i = 0; i < R_TUBES; ++i) {
      const float u = s[i];
      rank += (u > v) || (u == v && i < tid);
    }
    topk_out[rank] = v;
    end_idx[rank] = tid;
  }
  for (int r = R_TUBES + tid; r < KTOP; r += 256) {
    topk_out[r] = -1.0e30f;
    end_idx[r] = 0;
  }
}

// ---------------------------------------------------------------------------
// Phase 3: backtrack 500 chains through u8 backpointers.  Chunk the bp table
// into LDS (256 steps * 152 B = 38.9 KB) so the 1023 dependent lookups per
// chain hit LDS instead of L2.
// ---------------------------------------------------------------------------
#define CHUNK 256
__global__ void __launch_bounds__(512)
backtrack(const unsigned char* __restrict__ bp, const int* __restrict__ end_idx,
          unsigned char* __restrict__ pos) {
  __shared__ unsigned char tile[CHUNK][BP_STRIDE];
  const int tid = threadIdx.x;
  int idx = 0;
  if (tid < KTOP) {
    idx = end_idx[tid];
    pos[(size_t)tid * N_CLIPS + (N_CLIPS - 1)] = (unsigned char)idx;
  }
  for (int hi = N_CLIPS - 2; hi >= 0; hi -= CHUNK) {
    int lo = hi - CHUNK + 1; if (lo < 0) lo = 0;
    const int rows  = hi - lo + 1;
    const int words = rows * (BP_STRIDE / 4);
    const uint32_t* src = (const uint32_t*)(bp + (size_t)lo * BP_STRIDE);
    uint32_t* dst = (uint32_t*)&tile[0][0];
    for (int w = tid; w < words; w += 512) dst[w] = src[w];
    __syncthreads();
    if (tid < KTOP) {
      for (int s2 = hi; s2 >= lo; --s2) {
        idx = tile[s2 - lo][idx];
        pos[(size_t)tid * N_CLIPS + s2] = (unsigned char)idx;
      }
    }
    __syncthreads();
  }
}

// ---------------------------------------------------------------------------
// Phase 4: bandwidth phase.  out[k,t,0:64] = p_tubes[t, pos[k,t], 0:64].
// 16 threads per (k,t) pair, float4 per thread -> 256 B contiguous per pair.
// ---------------------------------------------------------------------------
__global__ void __launch_bounds__(256)
gather_out(const float* __restrict__ p_tubes, const unsigned char* __restrict__ pos,
           float* __restrict__ out) {
  const int tid  = threadIdx.x;
  const int p    = blockIdx.x * 16 + (tid >> 4);   // pair index k*1024 + t
  const int lane = tid & 15;
  const int t    = p & (N_CLIPS - 1);
  const int r    = pos[p];
  const float4* src = (const float4*)(p_tubes + ((size_t)t * R_TUBES + r) * D4) + lane;
  float4*       dst = (float4*)(out + (size_t)p * D4) + lane;
  *dst = *src;
}

// ---------------------------------------------------------------------------
extern "C" void kernel_launch(void* const* d_in, const int* in_sizes, int n_in,
                              void* d_out, int out_size, void* d_ws, size_t ws_size,
                              hipStream_t stream) {
  (void)in_sizes; (void)n_in; (void)out_size; (void)ws_size;
  const float* p_tubes = (const float*)d_in[0];
  const float* act     = (const float*)d_in[1];
  float* out = (float*)d_out;

  unsigned char* ws       = (unsigned char*)d_ws;
  unsigned char* bp       = ws + BP_OFF;
  float*         scores   = (float*)(ws + SC_OFF);
  int*           end_idx  = (int*)(ws + EI_OFF);
  unsigned char* pos      = ws + POS_OFF;
  float*         topk_out = out + (size_t)KTOP * N_CLIPS * D4;  // 32,768,000

  viterbi_fwd<<<1, 640, 0, stream>>>(p_tubes, act, bp, scores);
  topk_sort  <<<1, 256, 0, stream>>>(scores, topk_out, end_idx);
  backtrack  <<<1, 512, 0, stream>>>(bp, end_idx, pos);
  gather_out <<<(KTOP * N_CLIPS) / 16, 256, 0, stream>>>(p_tubes, pos, out);
}